// DeChunkLayerReference_50629074485307
// MI455X (gfx1250) — compile-verified
//
#include <hip/hip_runtime.h>
#include <math.h>

// Problem constants (match reference setup_inputs / globals)
#define BSZ    2
#define LFULL  4096
#define MSEQ   2048      // hidden seq length (M)
#define DMODEL 2048
#define NCHUNK 8         // MSEQ / BLK
#define BLK    256
#define EPSV   1e-4f
#define LDP    132       // padded LDS row pitch (floats): 132%64=4 banks/row shift

typedef __attribute__((ext_vector_type(2))) float v2f;
typedef __attribute__((ext_vector_type(4))) float v4f;
typedef __attribute__((ext_vector_type(8))) float v8f;
typedef __attribute__((ext_vector_type(4))) int   v4i;

#define AS1 __attribute__((address_space(1)))
#define AS3 __attribute__((address_space(3)))

// ---------------------------------------------------------------------------
// Kernel 1: per-batch serial prep (tiny). boundary_mask is numpy bool (1 byte).
// ---------------------------------------------------------------------------
__global__ void prep_kernel(const unsigned char* __restrict__ bmask,
                            const float* __restrict__ bprob,
                            float* __restrict__ coef, float* __restrict__ edec,
                            float* __restrict__ cl,   float* __restrict__ sdo,
                            int* __restrict__ rb, int* __restrict__ re,
                            int* __restrict__ sel)
{
    const int b = blockIdx.x;
    if (threadIdx.x != 0) return;
    const unsigned char* m  = bmask + (size_t)b * LFULL;
    const float*         pr = bprob + (size_t)b * LFULL * 2;
    int* s = sel + (size_t)b * LFULL;

    // stable partition: boundary tokens first, then the rest; take first MSEQ
    int pos = 0;
    for (int l = 0; l < LFULL; ++l) if (m[l]  && pos < MSEQ) s[pos++] = l;
    for (int l = 0; l < LFULL; ++l) if (!m[l] && pos < MSEQ) s[pos++] = l;

    float run = 0.f;
    for (int t = 0; t < MSEQ; ++t) {
        float p = pr[(size_t)s[t] * 2 + 1];
        p = fminf(fmaxf(p, EPSV), 1.f - EPSV);
        float dt = -logf(1.f - p);            // log(1/(1-p))
        coef[b * MSEQ + t] = p / dt;
        edec[b * MSEQ + t] = 1.f - p;         // exp(-dt) exactly
        if ((t & (BLK - 1)) == 0) run = 0.f;  // chunk-local cumsum
        run += -dt;
        cl [b * MSEQ + t] = run;
        sdo[b * MSEQ + t] = expf(run);
    }

    for (int t = 0; t < MSEQ; ++t) { rb[b * MSEQ + t] = 0; re[b * MSEQ + t] = 0; }
    int cum = 0, prev = -1;
    for (int l = 0; l < LFULL; ++l) {
        cum += m[l] ? 1 : 0;
        int t = cum - 1;
        t = t < 0 ? 0 : (t >= MSEQ ? MSEQ - 1 : t);
        if (t != prev) {
            rb[b * MSEQ + t] = l;
            if (prev >= 0) re[b * MSEQ + prev] = l;
            prev = t;
        }
    }
    if (prev >= 0) re[b * MSEQ + prev] = LFULL;
}

// ---------------------------------------------------------------------------
// Kernel 2: M[l,s] = exp(cl[l]-cl[s])*coef[s] for s<=l, else 0. 4 MB, L2-resident.
// ---------------------------------------------------------------------------
__global__ void buildM_kernel(const float* __restrict__ cl,
                              const float* __restrict__ coef,
                              float* __restrict__ Mbuf)
{
    const int bc = blockIdx.x;                 // b*NCHUNK + c
    const int b  = bc / NCHUNK, c = bc % NCHUNK;
    const int l  = threadIdx.x;
    const float* clc = cl   + b * MSEQ + c * BLK;
    const float* cfc = coef + b * MSEQ + c * BLK;
    const float  cll = clc[l];
    float* Mr = Mbuf + (size_t)bc * BLK * BLK + (size_t)l * BLK;
    for (int sidx = 0; sidx < BLK; ++sidx)
        Mr[sidx] = (sidx <= l) ? expf(cll - clc[sidx]) * cfc[sidx] : 0.f;
}

// ---------------------------------------------------------------------------
// Kernel 3: streaming inter-chunk scan -> chunk-entry states h_in[b,c,d].
// ---------------------------------------------------------------------------
__global__ void states_kernel(const float* __restrict__ hidden,
                              const float* __restrict__ edec,
                              const float* __restrict__ coef,
                              float* __restrict__ hin)
{
    const int b = blockIdx.x >> 3;
    const int d = ((blockIdx.x & 7) << 8) + threadIdx.x;
    float h = 0.f;
    for (int c = 0; c < NCHUNK; ++c) {
        hin[(size_t)(b * NCHUNK + c) * DMODEL + d] = h;
        for (int l = 0; l < BLK; ++l) {
            const int t = c * BLK + l;
            h = edec[b * MSEQ + t] * h +
                coef[b * MSEQ + t] * hidden[(size_t)(b * MSEQ + t) * DMODEL + d];
        }
    }
}

// ---------------------------------------------------------------------------
// Kernel 4: intra-chunk triangular GEMM (V_WMMA_F32_16X16X4_F32) with the
// block's 256x128 hidden stripe staged in LDS via async global->LDS DMA.
// Grid: BSZ*NCHUNK*16 blocks x 256 threads (8 waves); dynamic LDS = 256*LDP*4.
// ---------------------------------------------------------------------------
__global__ void __launch_bounds__(256)
ssd_wmma_kernel(const float* __restrict__ hidden,
                const float* __restrict__ Mbuf,
                const float* __restrict__ sdo,
                const float* __restrict__ hin,
                const int* __restrict__ rb, const int* __restrict__ re,
                float* __restrict__ out)
{
    extern __shared__ float tileH[];           // [256][LDP]

    const int blk  = blockIdx.x;
    const int dblk = blk & 15;                 // 16 d-blocks of 128
    const int c    = (blk >> 4) & 7;
    const int b    = blk >> 7;
    const int wave = threadIdx.x >> 5;         // 8 waves * 16 cols = 128 d
    const int lane = threadIdx.x & 31;
    const int half = lane >> 4;
    const int l16  = lane & 15;
    const int dbase = dblk * 128 + wave * 16;

    const float* Mb = Mbuf + (size_t)(b * NCHUNK + c) * BLK * BLK;
    const float* Hc = hidden + ((size_t)b * MSEQ + c * BLK) * DMODEL;

    // ---- stage hidden[cBLK..cBLK+255, dblk*128..+127] into LDS (16B chunks) ----
    // 256 rows * 128 cols / 4 floats = 8192 chunks; 32 per thread.
    for (int i = 0; i < 32; ++i) {
        const int j    = i * 256 + threadIdx.x;
        const int row  = j >> 5;               // 32 x 16B chunks per row
        const int col4 = (j & 31) << 2;
        const float* gp = Hc + (size_t)row * DMODEL + dblk * 128 + col4;
        float* lp = &tileH[row * LDP + col4];
#if __has_builtin(__builtin_amdgcn_global_load_async_to_lds_b128)
        __builtin_amdgcn_global_load_async_to_lds_b128(
            (AS1 v4i*)gp, (AS3 v4i*)lp, 0, 0);
#else
        *(v4f*)lp = *(const v4f*)gp;           // fallback: b128 load + ds_store
#endif
    }
#if __has_builtin(__builtin_amdgcn_global_load_async_to_lds_b128)
#if __has_builtin(__builtin_amdgcn_s_wait_asynccnt)
    __builtin_amdgcn_s_wait_asynccnt(0);
#else
    asm volatile("s_wait_asynccnt 0x0" ::: "memory");
#endif
#endif
    __syncthreads();

    const float hval = hin[(size_t)(b * NCHUNK + c) * DMODEL + dbase + l16];

    for (int lt = 0; lt < 16; ++lt) {
        v8f acc = {0.f, 0.f, 0.f, 0.f, 0.f, 0.f, 0.f, 0.f};
        const int kmax = (lt + 1) * 16;                    // causal K range
        const float* arow = Mb + (size_t)(lt * 16 + l16) * BLK + 2 * half;
        if (lt < 15)                                        // warm L2/WGP$ for next tile
            __builtin_prefetch(Mb + (size_t)((lt + 1) * 16 + l16) * BLK, 0, 1);
        for (int k = 0; k < kmax; k += 4) {
            // A 16x4 f32: lanes 0-15 -> K=k..k+1, lanes 16-31 -> K=k+2..k+3
            v2f a = *(const v2f*)(arow + k);
            // B 4x16 f32 from LDS: column = wave*16+l16, rows k+2*half, +1
            const int lcol = wave * 16 + l16;
            v2f bb;
            bb.x = tileH[(k + 2 * half)     * LDP + lcol];
            bb.y = tileH[(k + 2 * half + 1) * LDP + lcol];
            acc = __builtin_amdgcn_wmma_f32_16x16x4_f32(
                false, a, false, bb, (short)0, acc, false, false);
        }
        // epilogue: Y_off = sdo[l]*h_in[d]; replicate row into out rows [rb,re)
#pragma unroll
        for (int r = 0; r < 8; ++r) {
            const int lrow = lt * 16 + r + 8 * half;       // C/D layout: M=r / M=8+r
            const int t    = c * BLK + lrow;
            const float y  = acc[r] + sdo[b * MSEQ + t] * hval;
            const int beg = rb[b * MSEQ + t], end = re[b * MSEQ + t];
            for (int lo = beg; lo < end; ++lo)
                out[((size_t)b * LFULL + lo) * DMODEL + dbase + l16] = y;
        }
    }
}

// ---------------------------------------------------------------------------
// Host launcher
// ---------------------------------------------------------------------------
extern "C" void kernel_launch(void* const* d_in, const int* in_sizes, int n_in,
                              void* d_out, int out_size, void* d_ws, size_t ws_size,
                              hipStream_t stream)
{
    const float*         hidden = (const float*)d_in[0];          // (2,2048,2048) f32
    const unsigned char* bmask  = (const unsigned char*)d_in[1];  // (2,4096) bool
    const float*         bprob  = (const float*)d_in[2];          // (2,4096,2) f32
    // d_in[3] (mask) unused by the reference math.
    float* out = (float*)d_out;                                   // (2,4096,2048) f32

    // workspace layout (4-byte elements); total ~4.3 MB
    float* wsf  = (float*)d_ws;
    float* coef = wsf + 0;                         // 2*2048
    float* edec = coef + BSZ * MSEQ;               // 2*2048
    float* cl   = edec + BSZ * MSEQ;               // 2*2048
    float* sdo  = cl   + BSZ * MSEQ;               // 2*2048
    float* hin  = sdo  + BSZ * MSEQ;               // 2*8*2048
    int*   rb   = (int*)(hin + BSZ * NCHUNK * DMODEL);   // 2*2048
    int*   re   = rb + BSZ * MSEQ;                 // 2*2048
    int*   sel  = re + BSZ * MSEQ;                 // 2*4096
    float* Mbuf = (float*)(sel + BSZ * LFULL);     // 2*8*256*256 = 4 MB

    const size_t lds_bytes = (size_t)BLK * LDP * sizeof(float);   // 132 KB

    prep_kernel  <<<BSZ, 1, 0, stream>>>(bmask, bprob, coef, edec, cl, sdo, rb, re, sel);
    buildM_kernel<<<BSZ * NCHUNK, BLK, 0, stream>>>(cl, coef, Mbuf);
    states_kernel<<<BSZ * (DMODEL / 256), 256, 0, stream>>>(hidden, edec, coef, hin);
    ssd_wmma_kernel<<<BSZ * NCHUNK * 16, 256, lds_bytes, stream>>>(
        hidden, Mbuf, sdo, hin, rb, re, out);
}